// EmbLearnerKNN_7653631721519
// MI455X (gfx1250) — compile-verified
//
#include <hip/hip_runtime.h>
#include <math.h>

// ---------------------------------------------------------------------------
// MI455X (gfx1250) implementation of the EmbLearnerKNN forward pass.
// - GEMMs on the fp32 WMMA path: V_WMMA_F32_16X16X4_F32 (full precision).
// - Graph propagation: one wave32 per edge, float4 gather + f32 global atomics
//   (destination buffers are 25.6MB -> scatter lives in the 192MB L2).
// - All reductions are fixed-order (deterministic) two-stage trees.
// ---------------------------------------------------------------------------

#define N_NODES 50000
#define HID     128
#define FIN     256
#define TAU_INV 5.0f     // 1/0.2
#define EPSN    1e-8f
#define KC      64       // K-chunk staged through LDS in the GEMM

typedef __attribute__((ext_vector_type(2))) float v2f;
typedef __attribute__((ext_vector_type(8))) float v8f;

__device__ __forceinline__ float dot4(float4 a, float4 b) {
  return a.x * b.x + a.y * b.y + a.z * b.z + a.w * b.w;
}

// ---------------------------------------------------------------------------
// GEMM: Y[N,128] = X[N,K] @ W[K,128] (+ bias). 256 threads = 8 wave32;
// block owns 16 rows x 128 cols; wave w owns the 16x16 tile at col w*16.
// K staged via LDS in chunks of KC. Uses V_WMMA_F32_16X16X4_F32.
// VGPR layouts per CDNA5 ISA 7.12.2:
//   A 16x4  f32: lanes 0-15 -> {K=k,k+1}, lanes 16-31 -> {K=k+2,k+3}, M=lane%16
//   B 4x16  f32: vgpr v, half h -> K = k + v + 2h, N = lane%16
//   C 16x16 f32: c[v] -> M = v + 8*(lane>=16), N = lane%16
// ---------------------------------------------------------------------------
__global__ __launch_bounds__(256) void k_gemm(const float* __restrict__ X,
                                              const float* __restrict__ W,
                                              const float* __restrict__ bias,
                                              float* __restrict__ Y,
                                              int K, int addBias) {
  __shared__ float ldsA[16 * KC];
  __shared__ float ldsW[KC * HID];
  const int tid  = threadIdx.x;
  const int wave = tid >> 5;
  const int lane = tid & 31;
  const int half = lane >> 4;
  const int l16  = lane & 15;
  const int rowBase = blockIdx.x * 16;
  const int colBase = wave * 16;

  v8f acc = {};

  for (int k0 = 0; k0 < K; k0 += KC) {
    // stage A panel (16 x KC) and W panel (KC x 128) in LDS
    for (int i = tid; i < 16 * KC; i += 256) {
      int r = i / KC, c = i % KC;
      ldsA[i] = X[(size_t)(rowBase + r) * K + k0 + c];
    }
    for (int i = tid; i < KC * HID; i += 256) {
      int r = i / HID, c = i % HID;
      ldsW[i] = W[(size_t)(k0 + r) * HID + c];
    }
    // hint-prefetch the next streaming X panel (global_prefetch_b8)
    if (k0 + KC < K)
      __builtin_prefetch(&X[(size_t)(rowBase + (tid & 15)) * K + k0 + KC], 0, 1);
    __syncthreads();

    for (int kk = 0; kk < KC; kk += 4) {
      const int ka = kk + 2 * half;
      v2f a, b;
      a.x = ldsA[l16 * KC + ka];
      a.y = ldsA[l16 * KC + ka + 1];
      b.x = ldsW[ka * HID + colBase + l16];
      b.y = ldsW[(ka + 1) * HID + colBase + l16];
      acc = __builtin_amdgcn_wmma_f32_16x16x4_f32(
          /*neg_a=*/false, a, /*neg_b=*/false, b,
          /*c_mod=*/(short)0, acc, /*reuse_a=*/false, /*reuse_b=*/false);
    }
    __syncthreads();
  }

  const int c = colBase + l16;
  const float bv = addBias ? bias[c] : 0.0f;
#pragma unroll
  for (int v = 0; v < 8; ++v) {
    const int r = rowBase + v + 8 * half;
    Y[(size_t)r * HID + c] = acc[v] + bv;
  }
}

// ---------------------------------------------------------------------------
// Degrees and symmetric GCN norm (self-loops counted via init=1).
// ---------------------------------------------------------------------------
__global__ void k_deg_init(int* __restrict__ deg) {
  int i = blockIdx.x * blockDim.x + threadIdx.x;
  if (i < N_NODES) deg[i] = 1;  // self-loop
}
__global__ void k_deg(const int* __restrict__ dst, int E, int* __restrict__ deg) {
  int e = blockIdx.x * blockDim.x + threadIdx.x;
  if (e < E) atomicAdd(&deg[dst[e]], 1);
}
__global__ void k_dinv(const int* __restrict__ deg, float* __restrict__ dinv) {
  int i = blockIdx.x * blockDim.x + threadIdx.x;
  if (i < N_NODES) {
    float d = (float)deg[i];
    dinv[i] = d > 0.0f ? rsqrtf(d) : 0.0f;
  }
}

// acc[i,:] = dinv[i]^2 * h[i,:]   (self-loop contribution, no atomics)
__global__ void k_self_init(const float* __restrict__ h,
                            const float* __restrict__ dinv,
                            float* __restrict__ acc) {
  int t = blockIdx.x * blockDim.x + threadIdx.x;
  if (t >= N_NODES * HID / 4) return;
  int node = t / (HID / 4);
  float w = dinv[node] * dinv[node];
  float4 v = ((const float4*)h)[t];
  float4 o = make_float4(v.x * w, v.y * w, v.z * w, v.w * w);
  ((float4*)acc)[t] = o;
}

// one wave32 per edge: lane owns 4 consecutive features
__global__ __launch_bounds__(256) void k_edge(const float* __restrict__ h,
                                              const int* __restrict__ src,
                                              const int* __restrict__ dst,
                                              const float* __restrict__ dinv,
                                              float* __restrict__ acc, int E) {
  int gid = blockIdx.x * blockDim.x + threadIdx.x;
  int e = gid >> 5;
  int lane = gid & 31;
  if (e >= E) return;
  int s = src[e], d = dst[e];
  float w = dinv[s] * dinv[d];
  float4 v = ((const float4*)(h + (size_t)s * HID))[lane];
  float* o = acc + (size_t)d * HID + lane * 4;
  unsafeAtomicAdd(o + 0, v.x * w);   // global_atomic_add_f32
  unsafeAtomicAdd(o + 1, v.y * w);
  unsafeAtomicAdd(o + 2, v.z * w);
  unsafeAtomicAdd(o + 3, v.w * w);
}

// out = relu(acc + bias [+ base])
__global__ void k_epilogue(const float* __restrict__ acc,
                           const float* __restrict__ bias,
                           const float* __restrict__ base,
                           float* __restrict__ out, int addBase) {
  int t = blockIdx.x * blockDim.x + threadIdx.x;
  if (t >= N_NODES * HID) return;
  int c = t & (HID - 1);
  float v = acc[t] + bias[c];
  if (addBase) v += base[t];
  out[t] = fmaxf(v, 0.0f);
}

// out[i,:] = b[:] + (i==q ? w[:] : 0)    (one-hot query @ [1,HID] weight)
__global__ void k_qlinear(const float* __restrict__ w,
                          const float* __restrict__ b,
                          float* __restrict__ out,
                          const int* __restrict__ qptr) {
  int t = blockIdx.x * blockDim.x + threadIdx.x;
  if (t >= N_NODES * HID) return;
  int node = t >> 7;
  int c = t & (HID - 1);
  float v = b ? b[c] : 0.0f;
  if (node == *qptr) v += w[c];
  out[t] = v;
}

// attention fusion: out = softmax([a.wa, b.wb]) weighted sum. One wave/node.
__global__ __launch_bounds__(256) void k_fuse(const float* __restrict__ A,
                                              const float* __restrict__ B,
                                              const float* __restrict__ wa,
                                              const float* __restrict__ wb,
                                              float* __restrict__ out) {
  int gid = blockIdx.x * blockDim.x + threadIdx.x;
  int node = gid >> 5;
  int lane = gid & 31;
  if (node >= N_NODES) return;
  float4 a = ((const float4*)(A + (size_t)node * HID))[lane];
  float4 b = ((const float4*)(B + (size_t)node * HID))[lane];
  float4 va = ((const float4*)wa)[lane];
  float4 vb = ((const float4*)wb)[lane];
  float pa = dot4(a, va);
  float pb = dot4(b, vb);
#pragma unroll
  for (int m = 16; m >= 1; m >>= 1) {
    pa += __shfl_xor(pa, m, 32);
    pb += __shfl_xor(pb, m, 32);
  }
  float mx = fmaxf(pa, pb);
  float ea = expf(pa - mx), eb = expf(pb - mx);
  float inv = 1.0f / (ea + eb);
  float aa = ea * inv, ab = eb * inv;
  float4 o = make_float4(aa * a.x + ab * b.x, aa * a.y + ab * b.y,
                         aa * a.z + ab * b.z, aa * a.w + ab * b.w);
  ((float4*)(out + (size_t)node * HID))[lane] = o;
}

// ------------------------------ loss stage ---------------------------------
__global__ void k_mask_clear(float* __restrict__ mask) {
  int i = blockIdx.x * blockDim.x + threadIdx.x;
  if (i < N_NODES) mask[i] = 0.0f;
}
__global__ void k_mask_set(const int* __restrict__ pos, float* __restrict__ mask) {
  int t = threadIdx.x;
  if (t < 64) mask[pos[t]] = 1.0f;
}
__global__ void k_mask_clearq(float* __restrict__ mask, const int* __restrict__ qptr) {
  mask[*qptr] = 0.0f;
}

__device__ __forceinline__ float keep_hash(unsigned j) {
  // deterministic surrogate for jax.random.uniform(key2)[q, j] > 0.3
  unsigned h = j * 2654435761u ^ 0x9e3779b9u;
  h ^= h >> 16; h *= 0x85ebca6bu; h ^= h >> 13;
  float u = (float)(h & 0xFFFFFFu) * (1.0f / 16777216.0f);
  return u > 0.3f ? 1.0f : 0.0f;
}

// Builds query-row vectors: qd[0:128]=hf[q], qd[128:256]=ha[q],
// qd[256:512]=x=z_pert[q], qd[512..514]=clamped norms.
__global__ __launch_bounds__(128) void k_qvec(const float* __restrict__ hf,
                                              const float* __restrict__ ha,
                                              const int* __restrict__ qptr,
                                              float* __restrict__ qd) {
  __shared__ float red[128];
  int j = threadIdx.x;
  int q = *qptr;
  float hfq = hf[(size_t)q * HID + j];
  float haq = ha[(size_t)q * HID + j];
  qd[j] = hfq;
  qd[HID + j] = haq;
  // surrogate for jax.random.permutation(key1)[q]
  unsigned pq = ((unsigned)q * 2654435761u + 12345u) % N_NODES;
  float zp0 = hf[(size_t)pq * HID + j];
  float zp1 = ha[(size_t)pq * HID + j];
  float x0 = hfq + 0.2f * zp0 * keep_hash(j);
  float x1 = haq + 0.2f * zp1 * keep_hash(j + HID);
  qd[256 + j] = x0;
  qd[256 + HID + j] = x1;

  float vals[3] = {hfq * hfq, haq * haq, x0 * x0 + x1 * x1};
  float sums[3];
#pragma unroll
  for (int s = 0; s < 3; ++s) {
    red[j] = vals[s];
    __syncthreads();
    for (int k = 64; k >= 1; k >>= 1) {
      if (j < k) red[j] += red[j + k];
      __syncthreads();
    }
    sums[s] = red[0];
    __syncthreads();
  }
  if (j == 0) {
    qd[512] = fmaxf(sqrtf(sums[0]), EPSN);
    qd[513] = fmaxf(sqrtf(sums[1]), EPSN);
    qd[514] = fmaxf(sqrtf(sums[2]), EPSN);
  }
}

// Per-block partials: one wave per node (8 nodes/block), fixed-order combine.
// slots: 0-3 = sum exp(cos_k/tau), 4-7 = sum mask*cos_k/tau,
//        8 = sum cos_z, 9 = sum mask, 10 = cos_z at node q, 11 = pad
__global__ __launch_bounds__(256) void k_loss_partial(const float* __restrict__ hf,
                                                      const float* __restrict__ ha,
                                                      const float* __restrict__ mask,
                                                      const float* __restrict__ qd,
                                                      const int* __restrict__ qptr,
                                                      float* __restrict__ partials) {
  __shared__ float sh[8][12];
  int wave = threadIdx.x >> 5;
  int lane = threadIdx.x & 31;
  int node = blockIdx.x * 8 + wave;
  if (node < N_NODES) {
    float4 f  = ((const float4*)(hf + (size_t)node * HID))[lane];
    float4 g  = ((const float4*)(ha + (size_t)node * HID))[lane];
    float4 fq = ((const float4*)qd)[lane];
    float4 gq = ((const float4*)(qd + HID))[lane];
    float4 x0 = ((const float4*)(qd + 256))[lane];
    float4 x1 = ((const float4*)(qd + 256 + HID))[lane];
    float p[8];
    p[0] = dot4(f, f);   p[1] = dot4(g, g);
    p[2] = dot4(f, fq);  p[3] = dot4(g, gq);
    p[4] = dot4(g, fq);  p[5] = dot4(f, gq);
    p[6] = dot4(f, x0);  p[7] = dot4(g, x1);
#pragma unroll
    for (int m = 16; m >= 1; m >>= 1)
#pragma unroll
      for (int s = 0; s < 8; ++s) p[s] += __shfl_xor(p[s], m, 32);
    if (lane == 0) {
      float nhf = fmaxf(sqrtf(p[0]), EPSN);
      float nha = fmaxf(sqrtf(p[1]), EPSN);
      float qn1 = qd[512], qn2 = qd[513], qnx = qd[514];
      float c1 = p[2] / (nhf * qn1);              // cos(hf[q], hf[i])
      float c2 = p[3] / (nha * qn2);              // cos(ha[q], ha[i])
      float c3 = p[4] / (nha * qn1);              // cos(hf[q], ha[i])
      float c4 = p[5] / (nhf * qn2);              // cos(ha[q], hf[i])
      float cz = (p[6] + p[7]) / (fmaxf(sqrtf(p[0] + p[1]), EPSN) * qnx);
      float m = mask[node];
      sh[wave][0] = expf(c1 * TAU_INV);
      sh[wave][1] = expf(c2 * TAU_INV);
      sh[wave][2] = expf(c3 * TAU_INV);
      sh[wave][3] = expf(c4 * TAU_INV);
      sh[wave][4] = m * c1 * TAU_INV;
      sh[wave][5] = m * c2 * TAU_INV;
      sh[wave][6] = m * c3 * TAU_INV;
      sh[wave][7] = m * c4 * TAU_INV;
      sh[wave][8] = cz;
      sh[wave][9] = m;
      sh[wave][10] = (node == *qptr) ? cz : 0.0f;
      sh[wave][11] = 0.0f;
    }
  } else if (lane == 0) {
    for (int s = 0; s < 12; ++s) sh[wave][s] = 0.0f;
  }
  __syncthreads();
  if (threadIdx.x < 12) {
    float s = 0.0f;
    for (int wv = 0; wv < 8; ++wv) s += sh[wv][threadIdx.x];
    partials[(size_t)blockIdx.x * 12 + threadIdx.x] = s;
  }
}

#define LB 6250  // number of loss-partial blocks (50000/8)

__global__ __launch_bounds__(256) void k_loss_final(const float* __restrict__ partials,
                                                    float* __restrict__ out) {
  __shared__ float red[256];
  __shared__ float tot[12];
  int t = threadIdx.x;
  for (int s = 0; s < 12; ++s) {
    float v = 0.0f;
    for (int i = t; i < LB; i += 256) v += partials[(size_t)i * 12 + s];
    red[t] = v;
    __syncthreads();
    for (int k = 128; k >= 1; k >>= 1) {
      if (t < k) red[t] += red[t + k];
      __syncthreads();
    }
    if (t == 0) tot[s] = red[0];
    __syncthreads();
  }
  if (t == 0) {
    float nmask = fmaxf(tot[9], 1.0f);
    // masked_nll(exp(cos/tau)) = log(sum exp(cos/tau)) - mean_mask(cos/tau)
    float nll1 = logf(tot[0]) - tot[4] / nmask;
    float nll2 = logf(tot[1]) - tot[5] / nmask;
    float nll3 = logf(tot[2]) - tot[6] / nmask;
    float nll4 = logf(tot[3]) - tot[7] / nmask;
    float total_intra = 0.5f * (nll1 + nll2);
    float total_inter = 0.5f * (nll3 + nll4);
    float unsup = -logf(tot[10] / tot[8]);
    out[0] = total_intra + 0.5f * total_inter + 0.5f * unsup;  // LAM=ALPHA=0.5
  }
}

// ---------------------------------------------------------------------------
// Host orchestration
// ---------------------------------------------------------------------------
extern "C" void kernel_launch(void* const* d_in, const int* in_sizes, int n_in,
                              void* d_out, int out_size, void* d_ws, size_t ws_size,
                              hipStream_t stream) {
  // setup_inputs() dict order:
  const float* feats = (const float*)d_in[0];
  const int* ei      = (const int*)d_in[1];
  const int* eia     = (const int*)d_in[2];
  const int* qptr    = (const int*)d_in[3];
  const int* pos     = (const int*)d_in[4];
  const float* Wq0   = (const float*)d_in[5];
  const float* bq0   = (const float*)d_in[6];
  const float* Wq1   = (const float*)d_in[7];
  const float* bq1   = (const float*)d_in[8];
  const float* W0    = (const float*)d_in[9];
  const float* b0    = (const float*)d_in[10];
  const float* W1    = (const float*)d_in[11];
  const float* b1    = (const float*)d_in[12];
  const float* Wf0   = (const float*)d_in[13];
  const float* bf0   = (const float*)d_in[14];
  const float* Wf1   = (const float*)d_in[15];
  const float* bf1   = (const float*)d_in[16];
  const float* qatt0 = (const float*)d_in[17];
  const float* att0  = (const float*)d_in[18];
  const float* qatt1 = (const float*)d_in[19];
  const float* att1  = (const float*)d_in[20];
  const float* fqatt = (const float*)d_in[21];
  const float* fatt  = (const float*)d_in[22];
  const float* Wlq   = (const float*)d_in[23];
  const float* blq   = (const float*)d_in[24];
  const float* Wlf   = (const float*)d_in[25];
  const float* blf   = (const float*)d_in[26];

  const int E  = in_sizes[1] / 2;
  const int Ea = in_sizes[2] / 2;
  const int* src  = ei;        const int* dst  = ei + E;
  const int* srca = eia;       const int* dsta = eia + Ea;

  // workspace layout (floats): 12 big [N,HID] buffers + small tail (~308 MB)
  float* ws = (float*)d_ws;
  const size_t BIG = (size_t)N_NODES * HID;
  float* B[12];
  for (int i = 0; i < 12; ++i) B[i] = ws + i * BIG;
  float* ACC = B[5];
  size_t off = 12 * BIG;
  float* dinv  = ws + off; off += N_NODES;
  float* dinva = ws + off; off += N_NODES;
  float* maskf = ws + off; off += N_NODES;
  float* qd    = ws + off; off += 1024;
  float* parts = ws + off; off += (size_t)LB * 12;
  int* deg  = (int*)(ws + off); off += N_NODES;
  int* dega = (int*)(ws + off); off += N_NODES;

  const int NB_N  = (N_NODES + 255) / 256;
  const int NB_NH = (N_NODES * HID) / 256;        // 25000
  const int NB_V4 = (N_NODES * HID / 4 + 255) / 256;

  auto gemm = [&](const float* X, const float* W, const float* bias, float* Y, int K) {
    k_gemm<<<N_NODES / 16, 256, 0, stream>>>(X, W, bias, Y, K, bias != nullptr);
  };
  auto prop = [&](const float* h, const int* s, const int* d, const float* dv, int Ecnt) {
    k_self_init<<<NB_V4, 256, 0, stream>>>(h, dv, ACC);
    k_edge<<<(int)(((size_t)Ecnt * 32 + 255) / 256), 256, 0, stream>>>(h, s, d, dv, ACC, Ecnt);
  };
  auto epi = [&](const float* bias, const float* base, float* out_) {
    k_epilogue<<<NB_NH, 256, 0, stream>>>(ACC, bias, base, out_, base != nullptr);
  };
  auto fuse = [&](const float* A, const float* Bb, const float* wa, const float* wb, float* out_) {
    k_fuse<<<(N_NODES * 32 + 255) / 256, 256, 0, stream>>>(A, Bb, wa, wb, out_);
  };

  // ---- degrees / norms for both graphs
  k_deg_init<<<NB_N, 256, 0, stream>>>(deg);
  k_deg_init<<<NB_N, 256, 0, stream>>>(dega);
  k_deg<<<(E + 255) / 256, 256, 0, stream>>>(dst, E, deg);
  k_deg<<<(Ea + 255) / 256, 256, 0, stream>>>(dsta, Ea, dega);
  k_dinv<<<NB_N, 256, 0, stream>>>(deg, dinv);
  k_dinv<<<NB_N, 256, 0, stream>>>(dega, dinva);

  // ---- shared pre-propagation products
  gemm(feats, W0, nullptr, B[0], FIN);                          // XW0
  gemm(feats, Wlf, blf, B[1], FIN);                             // FL
  k_qlinear<<<NB_NH, 256, 0, stream>>>(Wlq, blq, B[2], qptr);   // QL
  fuse(B[2], B[1], fqatt, fatt, B[3]);                          // HF_
  gemm(B[3], Wf0, nullptr, B[4], HID);                          // HF_ @ Wf0
  k_qlinear<<<NB_NH, 256, 0, stream>>>(Wq0, nullptr, B[2], qptr); // HQpre

  // ---- layer 0, original graph
  prop(B[2], src, dst, dinv, E);   epi(bq0, nullptr, B[6]);     // Hq
  prop(B[0], src, dst, dinv, E);   epi(b0,  nullptr, B[7]);     // H
  fuse(B[6], B[7], qatt0, att0, B[8]);                          // HF0
  prop(B[4], src, dst, dinv, E);   epi(bf0, B[8], B[8]);        // HF
  // ---- layer 0, augmented graph
  prop(B[2], srca, dsta, dinva, Ea); epi(bq0, nullptr, B[9]);   // HqA
  prop(B[0], srca, dsta, dinva, Ea); epi(b0,  nullptr, B[10]);  // HA
  fuse(B[9], B[10], qatt0, att0, B[11]);                        // HFA0
  prop(B[4], srca, dsta, dinva, Ea); epi(bf0, B[11], B[11]);    // HFA

  // ---- layer 1, original graph
  gemm(B[6], Wq1, nullptr, B[0], HID);
  prop(B[0], src, dst, dinv, E);   epi(bq1, nullptr, B[1]);     // Hq1
  gemm(B[7], W1, nullptr, B[0], HID);
  prop(B[0], src, dst, dinv, E);   epi(b1,  nullptr, B[2]);     // H1
  fuse(B[1], B[2], qatt1, att1, B[3]);                          // HFX
  gemm(B[8], Wf1, nullptr, B[0], HID);
  prop(B[0], src, dst, dinv, E);   epi(bf1, B[3], B[3]);        // hf final
  // ---- layer 1, augmented graph
  gemm(B[9], Wq1, nullptr, B[0], HID);
  prop(B[0], srca, dsta, dinva, Ea); epi(bq1, nullptr, B[1]);   // HqA1
  gemm(B[10], W1, nullptr, B[0], HID);
  prop(B[0], srca, dsta, dinva, Ea); epi(b1,  nullptr, B[2]);   // HA1
  fuse(B[1], B[2], qatt1, att1, B[4]);                          // HFAX
  gemm(B[11], Wf1, nullptr, B[0], HID);
  prop(B[0], srca, dsta, dinva, Ea); epi(bf1, B[4], B[4]);      // h_aug final

  // ---- contrastive loss on (B[3], B[4])
  k_mask_clear<<<NB_N, 256, 0, stream>>>(maskf);
  k_mask_set<<<1, 64, 0, stream>>>(pos, maskf);
  k_mask_clearq<<<1, 1, 0, stream>>>(maskf, qptr);
  k_qvec<<<1, 128, 0, stream>>>(B[3], B[4], qptr, qd);
  k_loss_partial<<<LB, 256, 0, stream>>>(B[3], B[4], maskf, qd, qptr, parts);
  k_loss_final<<<1, 256, 0, stream>>>(parts, (float*)d_out);
}